// deep_gcn_35613868818502
// MI455X (gfx1250) — compile-verified
//
#include <hip/hip_runtime.h>
#include <math.h>

typedef __attribute__((ext_vector_type(2))) float v2f;
typedef __attribute__((ext_vector_type(8))) float v8f;

// ---------------------------------------------------------------------------
// Utility kernels
// ---------------------------------------------------------------------------
__global__ void zero_f32_kernel(float* __restrict__ p, int n) {
    int i = blockIdx.x * blockDim.x + threadIdx.x;
    if (i < n) p[i] = 0.0f;
}

__global__ void degree_kernel(const int* __restrict__ src, const int* __restrict__ dst,
                              float* __restrict__ outdeg, float* __restrict__ indeg, int e) {
    int i = blockIdx.x * blockDim.x + threadIdx.x;
    if (i < e) {
        atomicAdd(&outdeg[src[i]], 1.0f);
        atomicAdd(&indeg[dst[i]], 1.0f);
    }
}

__global__ void norm_kernel(float* __restrict__ outn, float* __restrict__ inn, int n) {
    int i = blockIdx.x * blockDim.x + threadIdx.x;
    if (i < n) {
        outn[i] = rsqrtf(fmaxf(outn[i], 1.0f));
        inn[i]  = rsqrtf(fmaxf(inn[i], 1.0f));
    }
}

// ---------------------------------------------------------------------------
// GEMM:  HW[r, :] = out_norm[r] * (H[r, :] @ W)   via V_WMMA_F32_16X16X4_F32
//   H: n_rows x K (row-major f32), W: K x NOUT (row-major f32).
// W is staged in LDS in a k-pair-interleaved layout:
//   Wlds[(k>>1)*LROW + n*2 + (k&1)]
// so a lane's B fragment (K = kh*2 + {0,1}, N = r) is one aligned ds_load_b64
// landing directly in an even-aligned VGPR pair (no repack moves before WMMA).
// LROW is padded by 32 floats when 2*NOUT is a multiple of 64 so the two lane
// halves (kh=0 -> banks 0..31, kh=1 -> banks 32..63) never conflict.
//
// One wave computes a 16-row x NOUT tile; block = 8 waves = 128 rows.
// A fragment (16x4 f32): lane holds row (lane&15); VGPR j holds K = (lane>>4)*2 + j
// C/D (16x16 f32): VGPR v, lanes 0-15 -> M=v, lanes 16-31 -> M=v+8; N = lane&15
// ---------------------------------------------------------------------------
template <int K, int NOUT>
__global__ __launch_bounds__(256)
void gemm_scale_wmma(const float* __restrict__ H, const float* __restrict__ W,
                     const float* __restrict__ out_norm, float* __restrict__ HW,
                     int n_rows) {
    constexpr int NT   = NOUT / 16;                              // 16-wide column tiles
    constexpr int LROW = 2 * NOUT + (((2 * NOUT) % 64 == 0) ? 32 : 0);
    __shared__ __align__(16) float Wlds[(K / 2) * LROW];

    // Stage W interleaved by k-pairs.
    for (int i = threadIdx.x; i < K * NOUT; i += 256) {
        const int k = i / NOUT;
        const int c = i - k * NOUT;
        Wlds[(k >> 1) * LROW + c * 2 + (k & 1)] = W[i];
    }
    __syncthreads();

    const int wave = threadIdx.x >> 5;
    const int lane = threadIdx.x & 31;
    const int row_base = (blockIdx.x * 8 + wave) * 16;
    if (row_base >= n_rows) return;

    const int r  = lane & 15;   // row (A) / col (B) within tile
    const int kh = lane >> 4;   // K half-select

    v8f acc[NT] = {};

    const float* hrow = H + (size_t)(row_base + r) * K + kh * 2;

#pragma unroll 4
    for (int kb = 0; kb < K; kb += 4) {
        const float2 av = *(const float2*)(hrow + kb);
        v2f a; a.x = av.x; a.y = av.y;
        const float* brow = &Wlds[((kb >> 1) + kh) * LROW + r * 2];
#pragma unroll
        for (int t = 0; t < NT; ++t) {
            const v2f b = *(const v2f*)(brow + t * 32);
            acc[t] = __builtin_amdgcn_wmma_f32_16x16x4_f32(
                false, a, false, b, (short)0, acc[t], false, false);
        }
    }

    // Fused out_norm row-scaling on store.
#pragma unroll
    for (int v = 0; v < 8; ++v) {
        const int row = row_base + v + kh * 8;
        const float s = out_norm[row];
#pragma unroll
        for (int t = 0; t < NT; ++t) {
            HW[(size_t)row * NOUT + t * 16 + r] = acc[t][v] * s;
        }
    }
}

// ---------------------------------------------------------------------------
// Scatter-add aggregation: agg[dst[e], :] += hw[src[e], :]
// One thread per (edge, 4-float chunk); NOUT/4 consecutive threads per edge
// so the float4 gathers of one edge row are contiguous.
// ---------------------------------------------------------------------------
template <int NOUT>
__global__ void scatter_add_kernel(const float* __restrict__ hw,
                                   const int* __restrict__ src,
                                   const int* __restrict__ dst,
                                   float* __restrict__ agg, int n_edges) {
    constexpr int CH = NOUT / 4;
    const int tid = blockIdx.x * blockDim.x + threadIdx.x;
    const int e = tid / CH;
    const int c = tid % CH;
    if (e >= n_edges) return;
    const int s = src[e];
    const int d = dst[e];
    const float4 v = *(const float4*)(hw + (size_t)s * NOUT + c * 4);
    float* p = agg + (size_t)d * NOUT + c * 4;
    atomicAdd(p + 0, v.x);
    atomicAdd(p + 1, v.y);
    atomicAdd(p + 2, v.z);
    atomicAdd(p + 3, v.w);
}

// ---------------------------------------------------------------------------
// Finalize: out[r, :] = (relu?)(agg[r, :] * in_norm[r])
// ---------------------------------------------------------------------------
template <int NOUT, bool RELU>
__global__ void finalize_kernel(const float* __restrict__ agg,
                                const float* __restrict__ in_norm,
                                float* __restrict__ out, int n_nodes) {
    constexpr int CH = NOUT / 4;
    const int i = blockIdx.x * blockDim.x + threadIdx.x;
    if (i >= n_nodes * CH) return;
    const int row = i / CH;
    const float s = in_norm[row];
    float4 v = *(const float4*)(agg + (size_t)i * 4);
    v.x *= s; v.y *= s; v.z *= s; v.w *= s;
    if (RELU) {
        v.x = fmaxf(v.x, 0.0f); v.y = fmaxf(v.y, 0.0f);
        v.z = fmaxf(v.z, 0.0f); v.w = fmaxf(v.w, 0.0f);
    }
    *(float4*)(out + (size_t)i * 4) = v;
}

// ---------------------------------------------------------------------------
// Host-side orchestration
// ---------------------------------------------------------------------------
extern "C" void kernel_launch(void* const* d_in, const int* in_sizes, int n_in,
                              void* d_out, int out_size, void* d_ws, size_t ws_size,
                              hipStream_t stream) {
    const float* features = (const float*)d_in[0];   // n x 128
    const int*   src      = (const int*)d_in[1];     // e
    const int*   dst      = (const int*)d_in[2];     // e
    const float* W0       = (const float*)d_in[3];   // 128 x 64
    const float* W1       = (const float*)d_in[4];   // 64 x 64
    const float* W2       = (const float*)d_in[5];   // 64 x 64
    const float* W3       = (const float*)d_in[6];   // 64 x 16
    float*       out      = (float*)d_out;           // n x 16

    const int n = in_sizes[0] / 128;
    const int e = in_sizes[1];

    float* ws   = (float*)d_ws;
    float* bufA = ws;                        // n*64  (GEMM output hW)
    float* bufB = bufA + (size_t)n * 64;     // n*64  (scatter accumulator)
    float* bufC = bufB + (size_t)n * 64;     // n*64  (current node features h)
    float* outn = bufC + (size_t)n * 64;     // n     (out-degree -> out_norm)
    float* inn  = outn + n;                  // n     (in-degree  -> in_norm)

    const int TB = 256;
    dim3 blk(TB);

    // --- degree normalization -------------------------------------------
    zero_f32_kernel<<<dim3((2 * n + TB - 1) / TB), blk, 0, stream>>>(outn, 2 * n);
    degree_kernel<<<dim3((e + TB - 1) / TB), blk, 0, stream>>>(src, dst, outn, inn, e);
    norm_kernel<<<dim3((n + TB - 1) / TB), blk, 0, stream>>>(outn, inn, n);

    const dim3 gemm_grid((n + 127) / 128);
    const dim3 sc64_grid((unsigned)(((size_t)e * 16 + TB - 1) / TB));
    const dim3 sc16_grid((unsigned)(((size_t)e * 4 + TB - 1) / TB));
    const dim3 fin64_grid((n * 16 + TB - 1) / TB);
    const dim3 fin16_grid((n * 4 + TB - 1) / TB);
    const dim3 z64_grid((n * 64 + TB - 1) / TB);

    // --- layer 0: 128 -> 64, relu ---------------------------------------
    gemm_scale_wmma<128, 64><<<gemm_grid, blk, 0, stream>>>(features, W0, outn, bufA, n);
    zero_f32_kernel<<<z64_grid, blk, 0, stream>>>(bufB, n * 64);
    scatter_add_kernel<64><<<sc64_grid, blk, 0, stream>>>(bufA, src, dst, bufB, e);
    finalize_kernel<64, true><<<fin64_grid, blk, 0, stream>>>(bufB, inn, bufC, n);

    // --- layer 1: 64 -> 64, relu ----------------------------------------
    gemm_scale_wmma<64, 64><<<gemm_grid, blk, 0, stream>>>(bufC, W1, outn, bufA, n);
    zero_f32_kernel<<<z64_grid, blk, 0, stream>>>(bufB, n * 64);
    scatter_add_kernel<64><<<sc64_grid, blk, 0, stream>>>(bufA, src, dst, bufB, e);
    finalize_kernel<64, true><<<fin64_grid, blk, 0, stream>>>(bufB, inn, bufC, n);

    // --- layer 2: 64 -> 64, relu ----------------------------------------
    gemm_scale_wmma<64, 64><<<gemm_grid, blk, 0, stream>>>(bufC, W2, outn, bufA, n);
    zero_f32_kernel<<<z64_grid, blk, 0, stream>>>(bufB, n * 64);
    scatter_add_kernel<64><<<sc64_grid, blk, 0, stream>>>(bufA, src, dst, bufB, e);
    finalize_kernel<64, true><<<fin64_grid, blk, 0, stream>>>(bufB, inn, bufC, n);

    // --- layer 3: 64 -> 16, no relu, output -----------------------------
    gemm_scale_wmma<64, 16><<<gemm_grid, blk, 0, stream>>>(bufC, W3, outn, bufA, n);
    zero_f32_kernel<<<dim3((n * 16 + TB - 1) / TB), blk, 0, stream>>>(out, n * 16);
    scatter_add_kernel<16><<<sc16_grid, blk, 0, stream>>>(bufA, src, dst, out, e);
    finalize_kernel<16, false><<<fin16_grid, blk, 0, stream>>>(out, inn, out, n);
}